// GenPhiUsingSubId_26388279067317
// MI455X (gfx1250) — compile-verified
//
#include <hip/hip_runtime.h>

typedef __attribute__((ext_vector_type(2))) float v2f;
typedef __attribute__((ext_vector_type(8))) float v8f;

#define NV0 10242
#define NV1 40962
#define NV2 163842
#define NSUB 65536
#define HDIM 256
#define Y0N (3 * NV0) /* 30726 */

// ---------------- workspace layout (float offsets) ----------------
static const size_t WS_H     = 0;                       // 256
static const size_t WS_Y0    = WS_H + 256;              // 30726
static const size_t WS_LIN   = WS_Y0 + Y0N;             // NV1*21 (shared lin0/lin1)
static const size_t WS_X1    = WS_LIN + (size_t)NV1*21; // NV1*3
static const size_t WS_X2    = WS_X1 + (size_t)NV1*3;   // NV2*3
static const size_t WS_XB    = WS_X2 + (size_t)NV2*3;   // NV2*3
static const size_t WS_P     = WS_XB + (size_t)NV2*3;   // NV2*8
static const size_t WS_Q     = WS_P + (size_t)NV2*8;    // NV2*8
static const size_t WS_STATS = WS_Q + (size_t)NV2*8;    // 5*16
static const size_t WS_PART  = WS_STATS + 80;           // 256*16

#define STATS_BLOCKS 256

// ================= GEMV1: h[256] = sub_id . W_sub[c,:] + b_sub =================
__global__ void gemv1_kernel(const float* __restrict__ sub,
                             const float* __restrict__ Wsub,
                             const float* __restrict__ bsub,
                             float* __restrict__ h) {
  __shared__ float red[256];
  const int c = blockIdx.x;
  const int t = threadIdx.x;
  const float4* s4 = (const float4*)sub;
  const float4* w4 = (const float4*)(Wsub + (size_t)c * NSUB);
  float acc = 0.f;
  for (int i = t; i < NSUB / 4; i += 256) {
    float4 a = s4[i], b = w4[i];
    acc += a.x * b.x + a.y * b.y + a.z * b.z + a.w * b.w;
  }
  red[t] = acc;
  __syncthreads();
  for (int s = 128; s > 0; s >>= 1) {
    if (t < s) red[t] += red[t + s];
    __syncthreads();
  }
  if (t == 0) h[c] = red[0] + bsub[c];
}

// ============ GEMV2 via f32 WMMA: y0[r] = sum_k h[k]*W_fc[r,k] + b_fc[r] ============
// A = 16 rows of W_fc (16x4 tile), B = h chunk broadcast across all 16 columns.
__global__ void gemv2_wmma_kernel(const float* __restrict__ h,
                                  const float* __restrict__ Wfc,
                                  const float* __restrict__ bfc,
                                  float* __restrict__ y) {
  const int lane = threadIdx.x;          // 0..31
  const bool hi = lane >= 16;
  const int m = lane & 15;
  const int r0 = blockIdx.x * 16;
  int r = r0 + m;
  if (r >= Y0N) r = Y0N - 1;             // clamp loads for tail tile
  const float* arow = Wfc + (size_t)r * HDIM;
  v8f c = {};
  for (int k0 = 0; k0 < HDIM; k0 += 4) {
    const int ka = k0 + (hi ? 2 : 0);
    v2f a; a.x = arow[ka]; a.y = arow[ka + 1];
    v2f b; b.x = h[ka];    b.y = h[ka + 1];    // same value in every column
    c = __builtin_amdgcn_wmma_f32_16x16x4_f32(false, a, false, b, (short)0, c,
                                              false, false);
  }
  // D: lane L, vgpr v -> M = v + (hi?8:0), N = L%16. Column 0 lanes store.
  if ((lane & 15) == 0) {
    const int mbase = hi ? 8 : 0;
#pragma unroll
    for (int v = 0; v < 8; ++v) {
      const int rr = r0 + mbase + v;
      if (rr < Y0N) y[rr] = c[v] + bfc[rr];
    }
  }
}

// ================= deterministic 2-pass BatchNorm statistics =================
__device__ __forceinline__ void block_reduce_store(float v, float* dst,
                                                   float* red) {
  const int t = threadIdx.x;
  red[t] = v;
  __syncthreads();
  for (int s = 128; s > 0; s >>= 1) {
    if (t < s) red[t] += red[t + s];
    __syncthreads();
  }
  if (t == 0) *dst = red[0];
  __syncthreads();
}

__global__ void bn_stats_part_kernel(const float* __restrict__ x, int n, int c,
                                     float* __restrict__ part) {
  __shared__ float red[256];
  const int t = threadIdx.x;
  float ls[8], lq[8];
#pragma unroll
  for (int j = 0; j < 8; ++j) { ls[j] = 0.f; lq[j] = 0.f; }
  for (int i = blockIdx.x * 256 + t; i < n; i += gridDim.x * 256) {
#pragma unroll
    for (int j = 0; j < 8; ++j) {
      if (j < c) {
        const float v = x[(size_t)i * c + j];
        ls[j] += v;
        lq[j] += v * v;
      }
    }
  }
  float* base = part + (size_t)blockIdx.x * 16;
#pragma unroll
  for (int j = 0; j < 8; ++j) {
    if (j < c) {  // c is uniform -> barrier-safe
      block_reduce_store(ls[j], base + j, red);
      block_reduce_store(lq[j], base + 8 + j, red);
    }
  }
}

__global__ void bn_stats_final_kernel(const float* __restrict__ part,
                                      float* __restrict__ stats) {
  __shared__ float red[256];
  const int t = threadIdx.x;
  for (int j = 0; j < 16; ++j) {
    red[t] = part[(size_t)t * 16 + j];  // STATS_BLOCKS == 256 == blockDim
    __syncthreads();
    for (int s = 128; s > 0; s >>= 1) {
      if (t < s) red[t] += red[t + s];
      __syncthreads();
    }
    if (t == 0) stats[j] = red[0];
    __syncthreads();
  }
}

// ============ upconv: lin[n,21] = lrelu(bn(x[n,3])) @ up_W.T + up_b ============
__global__ void upconv_lin_kernel(const float* __restrict__ x, int n,
                                  const float* __restrict__ stats,
                                  const float* __restrict__ g,
                                  const float* __restrict__ bb,
                                  const float* __restrict__ W,
                                  const float* __restrict__ wb,
                                  float* __restrict__ lin) {
  const int i = blockIdx.x * blockDim.x + threadIdx.x;
  if (i >= n) return;
  const float inv_n = 1.0f / (float)n;
  float v[3];
#pragma unroll
  for (int ch = 0; ch < 3; ++ch) {
    const float mean = stats[ch] * inv_n;
    const float var = stats[8 + ch] * inv_n - mean * mean;
    const float rs = __frsqrt_rn(var + 1e-5f);
    const float z = (x[(size_t)i * 3 + ch] - mean) * rs * g[ch] + bb[ch];
    v[ch] = z > 0.f ? z : 0.2f * z;
  }
#pragma unroll
  for (int j = 0; j < 21; ++j) {
    float acc = wb[j];
#pragma unroll
    for (int ch = 0; ch < 3; ++ch) acc += W[j * 3 + ch] * v[ch];
    lin[(size_t)i * 21 + j] = acc;
  }
}

// ==== scatter/mean with the exact flat-reshape semantics of the reference ====
// xo flat element t:  t < ntop*3   -> lin[top[t/3]*3 + t%3]
//                     else q=2(t-s) -> 0.5*(flatd[q] + flatd[q+1]),
//                     flatd[q] = lin[down[q/3]*3 + q%3]
__global__ void upgather_kernel(const float* __restrict__ lin,
                                const int* __restrict__ top,
                                const int* __restrict__ down, int ntop,
                                int ntotal, float* __restrict__ xo) {
  const int t = blockIdx.x * blockDim.x + threadIdx.x;
  const int total = ntotal * 3;
  if (t >= total) return;
  const int split = ntop * 3;
  if (t < split) {
    const int i = t / 3, ch = t - i * 3;
    xo[t] = lin[(size_t)top[i] * 3 + ch];
  } else {
    const int tt = t - split;
    const int q0 = 2 * tt, q1 = 2 * tt + 1;
    const float a = lin[(size_t)down[q0 / 3] * 3 + (q0 % 3)];
    const float b = lin[(size_t)down[q1 / 3] * 3 + (q1 % 3)];
    xo[t] = 0.5f * (a + b);
  }
}

// ================= elementwise bn + leaky-relu =================
__global__ void bnact_kernel(const float* __restrict__ x, int n, int c,
                             const float* __restrict__ stats,
                             const float* __restrict__ g,
                             const float* __restrict__ bb,
                             float* __restrict__ y) {
  const int t = blockIdx.x * blockDim.x + threadIdx.x;
  if (t >= n * c) return;
  const int ch = t % c;
  const float inv_n = 1.0f / (float)n;
  const float mean = stats[ch] * inv_n;
  const float var = stats[8 + ch] * inv_n - mean * mean;
  const float rs = __frsqrt_rn(var + 1e-5f);
  const float z = (x[t] - mean) * rs * g[ch] + bb[ch];
  y[t] = z > 0.f ? z : 0.2f * z;
}

// ===== one-ring conv as f32 WMMA GEMM: out[v, n] = mat[v,:] @ W[n,:] + b[n] =====
// mat[v, s*CIN+ch] = xb[neigh[7v+s], ch]; lane L owns vertex m = L%16, keeps its
// full gathered row (7*CIN floats) in registers and feeds KP/4 WMMA steps.
template <int CIN, int COUT>
__global__ void onering_wmma_kernel(const float* __restrict__ xb,
                                    const int* __restrict__ neigh,
                                    const float* __restrict__ W,
                                    const float* __restrict__ bias, int nverts,
                                    float* __restrict__ out) {
  constexpr int K = 7 * CIN;
  constexpr int KP = (K + 3) & ~3;
  const int lane = threadIdx.x;
  const bool hi = lane >= 16;
  const int m = lane & 15;
  const int r0 = blockIdx.x * 16;
  int vtx = r0 + m;
  if (vtx >= nverts) vtx = nverts - 1;  // clamp gathers for tail tile

  float row[KP];
#pragma unroll
  for (int s = 0; s < 7; ++s) {
    const int idx = neigh[7 * vtx + s];
#pragma unroll
    for (int ch = 0; ch < CIN; ++ch)
      row[s * CIN + ch] = xb[(size_t)idx * CIN + ch];
  }
#pragma unroll
  for (int k = K; k < KP; ++k) row[k] = 0.f;

  const int n = lane & 15;
  v8f c = {};
#pragma unroll
  for (int k0 = 0; k0 < KP; k0 += 4) {
    v2f a;
    a.x = hi ? row[k0 + 2] : row[k0];
    a.y = hi ? row[k0 + 3] : row[k0 + 1];
    const int kb0 = k0 + (hi ? 2 : 0);
    const int kb1 = kb0 + 1;
    v2f b;
    b.x = (n < COUT && kb0 < K) ? W[n * K + kb0] : 0.f;
    b.y = (n < COUT && kb1 < K) ? W[n * K + kb1] : 0.f;
    c = __builtin_amdgcn_wmma_f32_16x16x4_f32(false, a, false, b, (short)0, c,
                                              false, false);
  }
  if (n < COUT) {
    const float bv = bias[n];
    const int mbase = hi ? 8 : 0;
#pragma unroll
    for (int v = 0; v < 8; ++v) {
      const int rr = r0 + mbase + v;
      if (rr < nverts) out[(size_t)rr * COUT + n] = c[v] + bv;
    }
  }
}

// =============================== launcher ===============================
extern "C" void kernel_launch(void* const* d_in, const int* in_sizes, int n_in,
                              void* d_out, int out_size, void* d_ws,
                              size_t ws_size, hipStream_t stream) {
  const float* sub_id = (const float*)d_in[0];
  const float* W_sub  = (const float*)d_in[1];
  const float* b_sub  = (const float*)d_in[2];
  const float* W_fc   = (const float*)d_in[3];
  const float* b_fc   = (const float*)d_in[4];
  const float* bn_u0g = (const float*)d_in[5];
  const float* bn_u0b = (const float*)d_in[6];
  const float* up0_W  = (const float*)d_in[7];
  const float* up0_b  = (const float*)d_in[8];
  const float* bn_u1g = (const float*)d_in[9];
  const float* bn_u1b = (const float*)d_in[10];
  const float* up1_W  = (const float*)d_in[11];
  const float* up1_b  = (const float*)d_in[12];
  const float* bn0g   = (const float*)d_in[13];
  const float* bn0b   = (const float*)d_in[14];
  const float* c0_W   = (const float*)d_in[15];
  const float* c0_b   = (const float*)d_in[16];
  const float* bn1g   = (const float*)d_in[17];
  const float* bn1b   = (const float*)d_in[18];
  const float* c1_W   = (const float*)d_in[19];
  const float* c1_b   = (const float*)d_in[20];
  const float* bn2g   = (const float*)d_in[21];
  const float* bn2b   = (const float*)d_in[22];
  const float* c2_W   = (const float*)d_in[23];
  const float* c2_b   = (const float*)d_in[24];
  const int* neigh = (const int*)d_in[25];
  const int* top0  = (const int*)d_in[26];
  const int* down0 = (const int*)d_in[27];
  const int* top1  = (const int*)d_in[28];
  const int* down1 = (const int*)d_in[29];

  float* ws = (float*)d_ws;
  float* out = (float*)d_out;
  float* h = ws + WS_H;
  float* y0 = ws + WS_Y0;
  float* lin = ws + WS_LIN;
  float* x1 = ws + WS_X1;
  float* x2 = ws + WS_X2;
  float* xb = ws + WS_XB;
  float* P = ws + WS_P;
  float* Q = ws + WS_Q;
  float* stats = ws + WS_STATS;
  float* part = ws + WS_PART;

  // 1) h = sub_id @ W_sub.T + b_sub   (HBM-bound streaming GEMV, 67 MB)
  gemv1_kernel<<<HDIM, 256, 0, stream>>>(sub_id, W_sub, b_sub, h);

  // 2) y0 = h @ W_fc.T + b_fc         (f32 WMMA GEMV, 31.5 MB)
  const int tiles_y0 = (Y0N + 15) / 16;
  gemv2_wmma_kernel<<<tiles_y0, 32, 0, stream>>>(h, W_fc, b_fc, y0);

  // 3) level-0 upconv: V0 -> V1
  bn_stats_part_kernel<<<STATS_BLOCKS, 256, 0, stream>>>(y0, NV0, 3, part);
  bn_stats_final_kernel<<<1, 256, 0, stream>>>(part, stats + 0);
  upconv_lin_kernel<<<(NV0 + 255) / 256, 256, 0, stream>>>(
      y0, NV0, stats + 0, bn_u0g, bn_u0b, up0_W, up0_b, lin);
  upgather_kernel<<<(NV1 * 3 + 255) / 256, 256, 0, stream>>>(lin, top0, down0,
                                                             NV0, NV1, x1);

  // 4) level-1 upconv: V1 -> V2
  bn_stats_part_kernel<<<STATS_BLOCKS, 256, 0, stream>>>(x1, NV1, 3, part);
  bn_stats_final_kernel<<<1, 256, 0, stream>>>(part, stats + 16);
  upconv_lin_kernel<<<(NV1 + 255) / 256, 256, 0, stream>>>(
      x1, NV1, stats + 16, bn_u1g, bn_u1b, up1_W, up1_b, lin);
  upgather_kernel<<<(NV2 * 3 + 255) / 256, 256, 0, stream>>>(lin, top1, down1,
                                                             NV1, NV2, x2);

  const int tiles_v2 = (NV2 + 15) / 16;

  // 5) c0: 3 -> 8 (K = 21 padded to 24)
  bn_stats_part_kernel<<<STATS_BLOCKS, 256, 0, stream>>>(x2, NV2, 3, part);
  bn_stats_final_kernel<<<1, 256, 0, stream>>>(part, stats + 32);
  bnact_kernel<<<(NV2 * 3 + 255) / 256, 256, 0, stream>>>(x2, NV2, 3,
                                                          stats + 32, bn0g,
                                                          bn0b, xb);
  onering_wmma_kernel<3, 8><<<tiles_v2, 32, 0, stream>>>(xb, neigh, c0_W, c0_b,
                                                         NV2, P);

  // 6) c1: 8 -> 8 (K = 56)
  bn_stats_part_kernel<<<STATS_BLOCKS, 256, 0, stream>>>(P, NV2, 8, part);
  bn_stats_final_kernel<<<1, 256, 0, stream>>>(part, stats + 48);
  bnact_kernel<<<(NV2 * 8 + 255) / 256, 256, 0, stream>>>(P, NV2, 8,
                                                          stats + 48, bn1g,
                                                          bn1b, Q);
  onering_wmma_kernel<8, 8><<<tiles_v2, 32, 0, stream>>>(Q, neigh, c1_W, c1_b,
                                                         NV2, P);

  // 7) c2: 8 -> 2 (K = 56) -> d_out [163842, 2] fp32
  bn_stats_part_kernel<<<STATS_BLOCKS, 256, 0, stream>>>(P, NV2, 8, part);
  bn_stats_final_kernel<<<1, 256, 0, stream>>>(part, stats + 64);
  bnact_kernel<<<(NV2 * 8 + 255) / 256, 256, 0, stream>>>(P, NV2, 8,
                                                          stats + 64, bn2g,
                                                          bn2b, Q);
  onering_wmma_kernel<8, 2><<<tiles_v2, 32, 0, stream>>>(Q, neigh, c2_W, c2_b,
                                                         NV2, out);

  (void)in_sizes; (void)n_in; (void)out_size; (void)ws_size;
}